// CRDLoss_4492535792355
// MI455X (gfx1250) — compile-verified
//
#include <hip/hip_runtime.h>
#include <hip/hip_bf16.h>

// ---- problem constants (from reference) ----
#define B_C   256
#define D_C   128
#define N_C   500000
#define K_C   2048
#define KP1   2049           // K+1 contrast columns
#define INV_T 14.285714285714286f   // 1/0.07
#define MOM_C 0.5f
#define EPS_C 1e-7f

typedef __attribute__((ext_vector_type(2))) float v2f;
typedef __attribute__((ext_vector_type(8))) float v8f;

// ------------------------------------------------------------------
// Kernel 0: zero the scalar accumulators in workspace
// ws[0]=sum(out_v1 e-values), ws[1]=sum(out_v2), ws[2]=loss term sum
// ------------------------------------------------------------------
__global__ void crd_zero(float* __restrict__ sums) {
    if (threadIdx.x < 4) sums[threadIdx.x] = 0.0f;
}

// ------------------------------------------------------------------
// Kernel 1: gathered dot products via V_WMMA_F32_16X16X4_F32 + exp.
// grid = (17, B), block = 256 (8 waves). Wave handles one 16-row tile
// of one b, for BOTH views (same gather indices).
//   out_v2[b,k] = exp(dot(mem1[idx], teacher[b]) / T)
//   out_v1[b,k] = exp(dot(mem2[idx], student[b]) / T)
// ------------------------------------------------------------------
__global__ void __launch_bounds__(256)
crd_dot_exp(const float* __restrict__ mem1, const float* __restrict__ mem2,
            const float* __restrict__ s_emb, const float* __restrict__ t_emb,
            const int*   __restrict__ cidx,
            float* __restrict__ out1, float* __restrict__ out2,
            float* __restrict__ sums)
{
    __shared__ float sh_s[D_C];
    __shared__ float sh_t[D_C];
    const int b   = blockIdx.y;
    const int tid = threadIdx.x;
    if (tid < D_C) sh_s[tid]       = s_emb[(size_t)b * D_C + tid];
    else           sh_t[tid - D_C] = t_emb[(size_t)b * D_C + (tid - D_C)];
    __syncthreads();

    const int wave = tid >> 5;
    const int lane = tid & 31;
    const int tile = blockIdx.x * 8 + wave;
    if (tile >= 129) return;               // ceil(2049/16) = 129 tiles per b

    const int k0    = tile * 16;
    const int laneM = lane & 15;           // A-matrix row (M) this lane feeds
    const int koff  = (lane >> 4) * 2;     // K sub-offset: lanes 16-31 carry K=2,3
    const int k     = k0 + laneM;
    const int kk    = (k < KP1) ? k : (KP1 - 1);   // clamp tail; masked on output
    const long row  = (long)cidx[(size_t)b * KP1 + kk];

    const float* r1 = mem1 + row * D_C + koff;
    const float* r2 = mem2 + row * D_C + koff;

    v8f c1 = {};   // accum: mem1 x teacher  -> out_v2
    v8f c2 = {};   // accum: mem2 x student  -> out_v1
#pragma unroll
    for (int j = 0; j < 32; ++j) {         // 32 chained K=4 steps cover D=128
        v2f a1 = *(const v2f*)(r1 + 4 * j);
        v2f a2 = *(const v2f*)(r2 + 4 * j);
        const int d0 = 4 * j + koff;
        v2f bt; bt[0] = sh_t[d0]; bt[1] = sh_t[d0 + 1];
        v2f bs; bs[0] = sh_s[d0]; bs[1] = sh_s[d0 + 1];
        c1 = __builtin_amdgcn_wmma_f32_16x16x4_f32(false, a1, false, bt,
                                                   (short)0, c1, false, false);
        c2 = __builtin_amdgcn_wmma_f32_16x16x4_f32(false, a2, false, bs,
                                                   (short)0, c2, false, false);
    }

    // All N columns of C are identical; lane 0 holds rows 0-7, lane 16 rows 8-15.
    float p1 = 0.0f, p2 = 0.0f;
    if (laneM == 0) {
        const int rbase = k0 + (lane >> 4) * 8;
#pragma unroll
        for (int r = 0; r < 8; ++r) {
            const int kr = rbase + r;
            if (kr < KP1) {
                const float e2 = expf(c1[r] * INV_T);  // out_v2
                const float e1 = expf(c2[r] * INV_T);  // out_v1
                out2[(size_t)b * KP1 + kr] = e2;
                out1[(size_t)b * KP1 + kr] = e1;
                p1 += e1; p2 += e2;
            }
        }
    }
#pragma unroll
    for (int off = 16; off > 0; off >>= 1) {
        p1 += __shfl_down(p1, off, 32);
        p2 += __shfl_down(p2, off, 32);
    }
    if (lane == 0) {
        atomicAdd(&sums[0], p1);
        atomicAdd(&sums[1], p2);
    }
}

// ------------------------------------------------------------------
// Kernel 2: loss terms. z_v = mean(e)*N; p = e/z;
//   k==0: log(p/(p+noise)); k>0: log(K*pn/(p+noise)), noise=K*pn+eps.
// Grid-stride + block reduce + one atomicAdd per block into sums[2].
// ------------------------------------------------------------------
__global__ void __launch_bounds__(256)
crd_loss_terms(const float* __restrict__ out1, const float* __restrict__ out2,
               const float* __restrict__ sums, float* __restrict__ lossAcc)
{
    const float scale = (float)N_C / (float)(B_C * KP1);
    const float z1 = sums[0] * scale;
    const float z2 = sums[1] * scale;
    const float pn    = 1.0f / (float)N_C;
    const float noise = (float)K_C * pn + EPS_C;
    const float num0  = (float)K_C * pn;
    const int M = B_C * KP1;

    float acc = 0.0f;
    for (int i = blockIdx.x * blockDim.x + threadIdx.x; i < 2 * M;
         i += gridDim.x * blockDim.x) {
        const bool v2 = (i >= M);
        const int  j  = v2 ? (i - M) : i;
        const float e = v2 ? out2[j] : out1[j];
        const float z = v2 ? z2 : z1;
        const float p = e / z;
        const int  kcol = j % KP1;
        acc += (kcol == 0) ? logf(p / (p + noise)) : logf(num0 / (p + noise));
    }
#pragma unroll
    for (int off = 16; off > 0; off >>= 1) acc += __shfl_down(acc, off, 32);
    __shared__ float wsum[8];
    const int lane = threadIdx.x & 31, wave = threadIdx.x >> 5;
    if (lane == 0) wsum[wave] = acc;
    __syncthreads();
    if (threadIdx.x == 0) {
        float t = 0.0f;
#pragma unroll
        for (int w = 0; w < 8; ++w) t += wsum[w];
        atomicAdd(lossAcc, t);
    }
}

__global__ void crd_finalize(const float* __restrict__ lossAcc,
                             float* __restrict__ out) {
    out[0] = -lossAcc[0] / (float)B_C;
}

// ------------------------------------------------------------------
// Kernel 4: stream both memory banks into d_out (scalar b32: dst is
// only 4B-aligned because the loss scalar occupies d_out[0]).
// ------------------------------------------------------------------
__global__ void __launch_bounds__(256)
crd_copy(const float* __restrict__ m1, const float* __restrict__ m2,
         float* __restrict__ o1, float* __restrict__ o2)
{
    const size_t n = (size_t)N_C * D_C;
    for (size_t i = (size_t)blockIdx.x * blockDim.x + threadIdx.x; i < n;
         i += (size_t)gridDim.x * blockDim.x) {
        o1[i] = m1[i];
        o2[i] = m2[i];
    }
}

// ------------------------------------------------------------------
// Kernel 5: momentum update + renormalize the 256 selected rows.
// Reads ORIGINAL inputs (untouched), overwrites rows in the copy.
// One block (128 threads) per row.
// ------------------------------------------------------------------
__global__ void __launch_bounds__(128)
crd_update(const float* __restrict__ mem, const float* __restrict__ emb,
           const int* __restrict__ pos, float* __restrict__ dst)
{
    const int b = blockIdx.x;
    const int d = threadIdx.x;
    const long p = (long)pos[b];
    const float l = mem[p * D_C + d] * MOM_C +
                    emb[(size_t)b * D_C + d] * (1.0f - MOM_C);
    float s = l * l;
#pragma unroll
    for (int off = 16; off > 0; off >>= 1) s += __shfl_down(s, off, 32);
    __shared__ float ws[4];
    __shared__ float tot;
    if ((d & 31) == 0) ws[d >> 5] = s;
    __syncthreads();
    if (d == 0) tot = ws[0] + ws[1] + ws[2] + ws[3];
    __syncthreads();
    dst[p * D_C + d] = l / sqrtf(tot);
}

// ------------------------------------------------------------------
extern "C" void kernel_launch(void* const* d_in, const int* in_sizes, int n_in,
                              void* d_out, int out_size, void* d_ws, size_t ws_size,
                              hipStream_t stream) {
    const float* s_emb = (const float*)d_in[0];   // student_embed  (B,D)
    const float* t_emb = (const float*)d_in[1];   // teacher_embed  (B,D)
    const float* mem1  = (const float*)d_in[2];   // memory_v1      (N,D)
    const float* mem2  = (const float*)d_in[3];   // memory_v2      (N,D)
    const int*   pos   = (const int*)d_in[4];     // pos_idx        (B,)
    const int*   cidx  = (const int*)d_in[5];     // contrast_idx   (B,K+1)

    float* out = (float*)d_out;
    float* o1  = out + 1;                          // new_memory_v1
    float* o2  = out + 1 + (size_t)N_C * D_C;      // new_memory_v2

    float* wsf  = (float*)d_ws;
    float* sums = wsf;                             // [0],[1] = z sums, [2] = loss acc
    float* e1   = wsf + 16;                        // out_v1 e-values (B*KP1)
    float* e2   = e1 + (size_t)B_C * KP1;          // out_v2 e-values (B*KP1)

    crd_zero<<<1, 32, 0, stream>>>(sums);
    crd_dot_exp<<<dim3(17, B_C), 256, 0, stream>>>(mem1, mem2, s_emb, t_emb,
                                                   cidx, e1, e2, sums);
    crd_loss_terms<<<1024, 256, 0, stream>>>(e1, e2, sums, sums + 2);
    crd_finalize<<<1, 1, 0, stream>>>(sums + 2, out);

    crd_copy<<<8192, 256, 0, stream>>>(mem1, mem2, o1, o2);
    crd_update<<<B_C, 128, 0, stream>>>(mem1, s_emb, pos, o1);   // v1 <- student
    crd_update<<<B_C, 128, 0, stream>>>(mem2, t_emb, pos, o2);   // v2 <- teacher
}